// Model_15075335209780
// MI455X (gfx1250) — compile-verified
//
#include <hip/hip_runtime.h>
#include <math.h>

// Model dims (compile-time constants from the reference)
#define LL 12
#define NN 2048
#define KK 8
#define SS 4
#define HH 256
#define BB (2*NN)   // both encoders stacked: 4096 rows

typedef float v2f __attribute__((ext_vector_type(2)));
typedef float v8f __attribute__((ext_vector_type(8)));

// ---------------------------------------------------------------------------
// Wave-level f32 WMMA: accumulate As(16 x KD, LDS row-major) @ B(KD x ldn
// slice, global) into 16x64 accumulators at column nGlob.
// A 16x4 layout: lanes 0-15 -> K=k0,k0+1 ; lanes 16-31 -> K=k0+2,k0+3 (M=lane&15)
// B  4x16 layout: VGPR0 rows k0 / k0+2, VGPR1 rows k0+1 / k0+3 (N=lane&15)
// ---------------------------------------------------------------------------
template<int KD>
__device__ __forceinline__ void wmma_acc_16x64(
    const float* __restrict__ As, const float* __restrict__ Bw, int ldn,
    int nGlob, v8f acc[4])
{
  const int lane = threadIdx.x & 31;
  const int ln   = lane & 15;
  const int hi   = lane >> 4;
  #pragma unroll 4
  for (int k0 = 0; k0 < KD; k0 += 4) {
    v2f a;
    a.x = As[ln*KD + k0 + 2*hi + 0];
    a.y = As[ln*KD + k0 + 2*hi + 1];
    #pragma unroll
    for (int t = 0; t < 4; ++t) {
      const int nn = nGlob + t*16 + ln;
      v2f b;
      b.x = Bw[(size_t)(k0 + 2*hi + 0)*ldn + nn];
      b.y = Bw[(size_t)(k0 + 2*hi + 1)*ldn + nn];
      acc[t] = __builtin_amdgcn_wmma_f32_16x16x4_f32(
          false, a, false, b, (short)0, acc[t], false, false);
    }
  }
}

__device__ __forceinline__ void zero_acc(v8f acc[4]) {
  #pragma unroll
  for (int t = 0; t < 4; ++t) {
    #pragma unroll
    for (int i = 0; i < 8; ++i) acc[t][i] = 0.0f;
  }
}

__device__ __forceinline__ float sigf(float x) { return 1.0f/(1.0f + expf(-x)); }

// ---------------------------------------------------------------------------
// Kernel 1: x[l,b,:] = gatheredEmb(l,b) @ lin_W + lin_b
// rows m = l*BB + b over L*B = 49152 ; Kdim = S*H = 1024 ; Ncols = 256
// A tile (16 x 1024) gathered from E in two 512-wide LDS chunks.
// ---------------------------------------------------------------------------
__global__ void __launch_bounds__(128)
embed_lin_kernel(const int* __restrict__ tok1, const int* __restrict__ tok2,
                 const float* __restrict__ E,  const float* __restrict__ linW,
                 const float* __restrict__ linb, float* __restrict__ x)
{
  __shared__ float As[16*512];
  __shared__ int   tokc[16*SS];
  const int m0  = blockIdx.x * 16;
  const int tid = threadIdx.x;

  if (tid < 16*SS) {
    const int r = tid >> 2, s = tid & 3;
    const int m = m0 + r;
    const int l = m / BB, b = m % BB;
    const int* tp = (b < NN) ? tok1 : tok2;
    const int  n  = (b < NN) ? b : b - NN;
    tokc[tid] = tp[((size_t)l*NN + n)*SS + s];
  }
  __syncthreads();

  const int wave = tid >> 5;
  v8f acc[4]; zero_acc(acc);

  for (int chunk = 0; chunk < 2; ++chunk) {
    __syncthreads();
    // stage gathered embedding rows: cols [512*chunk, 512*chunk+512)
    for (int q = tid; q < 16*128; q += 128) {
      const int r   = q >> 7;
      const int c4  = (q & 127) * 4;
      const int col = chunk*512 + c4;       // 0..1023
      const int s   = col >> 8;
      const int h   = col & 255;
      const float4 v = *(const float4*)&E[(size_t)tokc[r*SS + s]*HH + h];
      *(float4*)&As[r*512 + c4] = v;
    }
    __syncthreads();
    wmma_acc_16x64<512>(As, linW + (size_t)chunk*512*HH, HH, wave*64, acc);
  }

  // store D + bias
  const int lane = tid & 31, ln = lane & 15, hi = lane >> 4;
  #pragma unroll
  for (int t = 0; t < 4; ++t) {
    const int nn = wave*64 + t*16 + ln;
    const float bv = linb[nn];
    #pragma unroll
    for (int i = 0; i < 8; ++i)
      x[(size_t)(m0 + i + 8*hi)*HH + nn] = acc[t][i] + bv;
  }
}

// ---------------------------------------------------------------------------
// Kernel 2 (generic): C = A(16-row tiles, K=256) @ Bw(256 x ldn) + bias,
// optional tanh epilogue. grid.y covers 256-col chunks.
// ---------------------------------------------------------------------------
__global__ void __launch_bounds__(128)
gemm256_kernel(const float* __restrict__ A, const float* __restrict__ Bw,
               const float* __restrict__ bias, float* __restrict__ C,
               int ldn, int applyTanh)
{
  __shared__ float As[16*256];
  const int m0  = blockIdx.x * 16;
  const int tid = threadIdx.x;
  for (int q = tid; q < 16*64; q += 128) {
    const int r = q >> 6, c4 = (q & 63) * 4;
    *(float4*)&As[r*256 + c4] = *(const float4*)&A[(size_t)(m0 + r)*256 + c4];
  }
  __syncthreads();

  const int wave  = tid >> 5;
  const int nGlob = blockIdx.y*256 + wave*64;
  v8f acc[4]; zero_acc(acc);
  wmma_acc_16x64<256>(As, Bw, ldn, nGlob, acc);

  const int lane = tid & 31, ln = lane & 15, hi = lane >> 4;
  #pragma unroll
  for (int t = 0; t < 4; ++t) {
    const int nn = nGlob + t*16 + ln;
    const float bv = bias[nn];
    #pragma unroll
    for (int i = 0; i < 8; ++i) {
      float v = acc[t][i] + bv;
      if (applyTanh) v = tanhf(v);
      C[(size_t)(m0 + i + 8*hi)*ldn + nn] = v;
    }
  }
}

// ---------------------------------------------------------------------------
// Kernel 3: h_sum[b,:] = sum_k valid h_prev[child]   (one block per b)
// ---------------------------------------------------------------------------
__global__ void __launch_bounds__(256)
hsum_kernel(const int* __restrict__ idx1, const int* __restrict__ idx2,
            const float* __restrict__ hprev, float* __restrict__ hsum, int l)
{
  const int b   = blockIdx.x;
  const int col = threadIdx.x;
  const int enc = b >> 11;
  const int n   = b & (NN - 1);
  const int* ip = (enc ? idx2 : idx1) + ((size_t)l*NN + n)*KK;
  float s = 0.0f;
  #pragma unroll
  for (int k = 0; k < KK; ++k) {
    const int j = ip[k];
    if (j >= 1) s += hprev[((size_t)(enc*NN + j - 1))*HH + col];
  }
  hsum[(size_t)b*HH + col] = s;
}

// ---------------------------------------------------------------------------
// Kernel 4: fused forget-gate GEMM + reduction.
// Rows m = b*8 + k over B*K = 32768. A rows gathered = h_prev[child] (or 0).
// D layout puts the 8 k-rows of one node in-lane across the 8 acc elements:
//   bf[b,col] = sum_i sigmoid(D[i] + Wf[b,col] + Uf_b[col]) * c_child[i][col]
// ---------------------------------------------------------------------------
__global__ void __launch_bounds__(128)
fgate_kernel(const int* __restrict__ idx1, const int* __restrict__ idx2,
             const float* __restrict__ hprev, const float* __restrict__ cprev,
             const float* __restrict__ UfW, const float* __restrict__ Ufb,
             const float* __restrict__ Wx, float* __restrict__ bf, int l)
{
  __shared__ float As[16*256];
  __shared__ float Cs[16*256];
  __shared__ int   rowsrc[16];
  const int m0  = blockIdx.x * 16;
  const int tid = threadIdx.x;

  if (tid < 16) {
    const int m = m0 + tid;
    const int b = m >> 3, k = m & 7;
    const int enc = b >> 11, n = b & (NN - 1);
    const int* ip = (enc ? idx2 : idx1);
    const int j = ip[((size_t)l*NN + n)*KK + k];
    rowsrc[tid] = (j >= 1) ? (enc*NN + j - 1) : -1;
  }
  __syncthreads();

  for (int q = tid; q < 16*64; q += 128) {
    const int r = q >> 6, c4 = (q & 63) * 4;
    const int src = rowsrc[r];
    float4 hv = {0.f, 0.f, 0.f, 0.f};
    float4 cv = {0.f, 0.f, 0.f, 0.f};
    if (src >= 0) {
      hv = *(const float4*)&hprev[(size_t)src*HH + c4];
      cv = *(const float4*)&cprev[(size_t)src*HH + c4];
    }
    *(float4*)&As[r*256 + c4] = hv;
    *(float4*)&Cs[r*256 + c4] = cv;
  }
  __syncthreads();

  const int wave = tid >> 5;
  const int lane = tid & 31, ln = lane & 15, hi = lane >> 4;
  v8f acc[4]; zero_acc(acc);
  wmma_acc_16x64<256>(As, UfW, HH, wave*64, acc);

  const int b = (m0 >> 3) + hi;   // node for this lane half
  #pragma unroll
  for (int t = 0; t < 4; ++t) {
    const int col = wave*64 + t*16 + ln;
    const float wf = Wx[(size_t)b*(4*HH) + col];   // Wf slice of Wx
    const float ub = Ufb[col];
    float s = 0.0f;
    #pragma unroll
    for (int i = 0; i < 8; ++i) {
      const float f = sigf(acc[t][i] + wf + ub);
      s += f * Cs[(hi*8 + i)*256 + col];
    }
    bf[(size_t)b*HH + col] = s;
  }
}

// ---------------------------------------------------------------------------
// Kernel 5: gate combine -> (h_next, c_next)
// ---------------------------------------------------------------------------
__global__ void __launch_bounds__(256)
combine_kernel(const float* __restrict__ Wx, const float* __restrict__ iuo,
               const float* __restrict__ bf, float* __restrict__ hnext,
               float* __restrict__ cnext)
{
  const size_t g = (size_t)blockIdx.x*256 + threadIdx.x;   // over B*H
  const int b = (int)(g >> 8);
  const int col = (int)(g & 255);
  const float* wxr = Wx  + (size_t)b*(4*HH);
  const float* ir  = iuo + (size_t)b*(3*HH);
  const float iG = sigf(ir[col]         + wxr[HH   + col]);
  const float uG = tanhf(ir[HH + col]   + wxr[2*HH + col]);
  const float oG = sigf(ir[2*HH + col]  + wxr[3*HH + col]);
  const float nc = iG*uG + bf[g];
  hnext[g] = oG * tanhf(nc);
  cnext[g] = nc;
}

__global__ void __launch_bounds__(256)
zero_kernel(float* __restrict__ p, int n)
{
  const int i = blockIdx.x*256 + threadIdx.x;
  if (i < n) p[i] = 0.0f;
}

// ---------------------------------------------------------------------------
// Kernel 6: ad[n,:] = | (h1 + x1[L-1]) - (h2 + x2[L-1]) |
// ---------------------------------------------------------------------------
__global__ void __launch_bounds__(256)
absdiff_kernel(const float* __restrict__ hlast, const float* __restrict__ x,
               float* __restrict__ ad)
{
  const size_t g = (size_t)blockIdx.x*256 + threadIdx.x;   // over N*H
  const int n = (int)(g >> 8), col = (int)(g & 255);
  const float* xl = x + (size_t)(LL - 1)*BB*HH;
  const float a = hlast[(size_t)n*HH + col]        + xl[(size_t)n*HH + col];
  const float c = hlast[(size_t)(NN + n)*HH + col] + xl[(size_t)(NN + n)*HH + col];
  ad[g] = fabsf(a - c);
}

// ---------------------------------------------------------------------------
// Kernel 7: logits = xd @ out_W + out_b ; softmax. One wave per row.
// ---------------------------------------------------------------------------
__global__ void __launch_bounds__(256)
head_kernel(const float* __restrict__ xd, const float* __restrict__ outW,
            const float* __restrict__ outb, float* __restrict__ out)
{
  const int wave = threadIdx.x >> 5;
  const int lane = threadIdx.x & 31;
  const int row  = blockIdx.x*8 + wave;
  const float* xr = xd + (size_t)row*512;
  float s0 = 0.f, s1 = 0.f;
  for (int k = lane; k < 512; k += 32) {
    const float v = xr[k];
    s0 += v * outW[k*2 + 0];
    s1 += v * outW[k*2 + 1];
  }
  for (int off = 16; off; off >>= 1) {
    s0 += __shfl_down(s0, off, 32);
    s1 += __shfl_down(s1, off, 32);
  }
  if (lane == 0) {
    const float l0 = s0 + outb[0], l1 = s1 + outb[1];
    const float m  = fmaxf(l0, l1);
    const float e0 = expf(l0 - m), e1 = expf(l1 - m);
    const float d  = e0 + e1;
    out[(size_t)row*2 + 0] = e0 / d;
    out[(size_t)row*2 + 1] = e1 / d;
  }
}

// ---------------------------------------------------------------------------
extern "C" void kernel_launch(void* const* d_in, const int* in_sizes, int n_in,
                              void* d_out, int out_size, void* d_ws, size_t ws_size,
                              hipStream_t stream)
{
  const int*   tokens1 = (const int*)  d_in[0];
  const int*   idx1    = (const int*)  d_in[1];
  const int*   tokens2 = (const int*)  d_in[2];
  const int*   idx2    = (const int*)  d_in[3];
  const float* E       = (const float*)d_in[4];
  const float* linW    = (const float*)d_in[5];
  const float* linb    = (const float*)d_in[6];
  const float* UfW     = (const float*)d_in[7];
  const float* Ufb     = (const float*)d_in[8];
  const float* UiuoW   = (const float*)d_in[9];
  const float* Uiuob   = (const float*)d_in[10];
  const float* WW      = (const float*)d_in[11];
  const float* Wb      = (const float*)d_in[12];
  const float* denseW  = (const float*)d_in[13];
  const float* denseb  = (const float*)d_in[14];
  const float* outW    = (const float*)d_in[15];
  const float* outb    = (const float*)d_in[16];
  float* out = (float*)d_out;

  // Workspace carve-out (~111 MB of f32)
  float* ws   = (float*)d_ws;
  float* x    = ws; ws += (size_t)LL*BB*HH;   // 12.58M  input linear for all steps
  float* hA   = ws; ws += (size_t)BB*HH;      // h/c double buffers (hA,cA contiguous)
  float* cA   = ws; ws += (size_t)BB*HH;
  float* hB   = ws; ws += (size_t)BB*HH;
  float* cB   = ws; ws += (size_t)BB*HH;
  float* Wx   = ws; ws += (size_t)BB*4*HH;
  float* hsum = ws; ws += (size_t)BB*HH;
  float* bf   = ws; ws += (size_t)BB*HH;
  float* iuo  = ws; ws += (size_t)BB*3*HH;
  float* ad   = ws; ws += (size_t)NN*HH;
  float* xd   = ws; ws += (size_t)NN*512;

  // 1) embedding gather + input linear, all steps, both encoders
  embed_lin_kernel<<<dim3((LL*BB)/16), 128, 0, stream>>>(tokens1, tokens2, E, linW, linb, x);

  // 2) zero initial h/c (hA,cA contiguous)
  zero_kernel<<<dim3((2*BB*HH)/256), 256, 0, stream>>>(hA, 2*BB*HH);

  float* hp = hA; float* cp = cA; float* hn = hB; float* cn = cB;
  for (int l = 0; l < LL; ++l) {
    gemm256_kernel<<<dim3(BB/16, 4), 128, 0, stream>>>(
        x + (size_t)l*BB*HH, WW, Wb, Wx, 4*HH, 0);
    hsum_kernel<<<dim3(BB), 256, 0, stream>>>(idx1, idx2, hp, hsum, l);
    fgate_kernel<<<dim3((BB*KK)/16), 128, 0, stream>>>(
        idx1, idx2, hp, cp, UfW, Ufb, Wx, bf, l);
    gemm256_kernel<<<dim3(BB/16, 3), 128, 0, stream>>>(
        hsum, UiuoW, Uiuob, iuo, 3*HH, 0);
    combine_kernel<<<dim3((BB*HH)/256), 256, 0, stream>>>(Wx, iuo, bf, hn, cn);
    float* t;
    t = hp; hp = hn; hn = t;
    t = cp; cp = cn; cn = t;
  }

  // 3) head: |h1-h2| -> dense+tanh -> 2-way softmax
  absdiff_kernel<<<dim3((NN*HH)/256), 256, 0, stream>>>(hp, x, ad);
  gemm256_kernel<<<dim3(NN/16, 2), 128, 0, stream>>>(ad, denseW, denseb, xd, 512, 1);
  head_kernel<<<dim3(NN/8), 256, 0, stream>>>(xd, outW, outb, out);
}